// LogSqrt2Quantizer_32117765439888
// MI455X (gfx1250) — compile-verified
//
#include <hip/hip_runtime.h>

// Elementwise log-sqrt2 quant/dequant over 8x16x1024x1024 fp32 (N = 134,217,728).
// Bandwidth-bound: ~1.07 GB traffic -> ~46 us roofline at 23.3 TB/s HBM.
// Strategy: 4x-unrolled grid-stride of float4 (B128) non-temporal streaming
// loads/stores (working set >> 192MB L2, zero reuse), 32-bit indexing,
// integer-exact strength-reduced math, wave32.

typedef float v4f __attribute__((ext_vector_type(4)));

__device__ __forceinline__ float qdq_elem(float x, float inv_s, float s255) {
    // x_int = round(x/s); x_int = round(x_int + 1) -> int32   (RNE, matches jnp.round)
    int xi = (int)__builtin_rintf(__builtin_rintf(x * inv_s) + 1.0f);
    xi = xi < 1 ? 1 : xi;                       // reference guarantees >= 1; guard clz
    int e  = 31 - __clz(xi);                    // floor(log2(xi))
    int p  = 1 << e;                            // 2^e
    // dequant: floor(2^e + frac*2^(e-1)), frac in {0,1}; integer-exact incl. e==0:
    int dq = p + ((((xi - p) << 1) >= p) ? (p >> 1) : 0);
    int d  = (dq - 1) / 255;                    // subtract INT_BIAS, floor-div (dq-1 >= 0)
    d = d < 0 ? 0 : (d > 255 ? 255 : d);        // clip(., 0, 255)
    return (float)d * s255;                     // * 255 * s folded into one scale
}

__device__ __forceinline__ v4f qdq_vec4(v4f v, float inv_s, float s255) {
    v4f r;
    r.x = qdq_elem(v.x, inv_s, s255);
    r.y = qdq_elem(v.y, inv_s, s255);
    r.z = qdq_elem(v.z, inv_s, s255);
    r.w = qdq_elem(v.w, inv_s, s255);
    return r;
}

__global__ void __launch_bounds__(256)
logsqrt2_qdq_kernel(const float* __restrict__ x, const float* __restrict__ s_x,
                    float* __restrict__ out, unsigned n) {
    const float s     = s_x[0];
    const float inv_s = 1.0f / s;               // one divide/thread, SGPR-broadcast
    const float s255  = 255.0f * s;

    const unsigned n4     = n >> 2;
    const unsigned stride = gridDim.x * blockDim.x;
    unsigned       i      = blockIdx.x * blockDim.x + threadIdx.x;

    const v4f* __restrict__ x4 = (const v4f*)x;
    v4f* __restrict__       o4 = (v4f*)out;

    // 4x unrolled: 4 independent B128 NT loads in flight per iteration
    for (; i + 3u * stride < n4; i += 4u * stride) {
        v4f a = __builtin_nontemporal_load(&x4[i]);
        v4f b = __builtin_nontemporal_load(&x4[i + stride]);
        v4f c = __builtin_nontemporal_load(&x4[i + 2u * stride]);
        v4f d = __builtin_nontemporal_load(&x4[i + 3u * stride]);
        v4f ra = qdq_vec4(a, inv_s, s255);
        v4f rb = qdq_vec4(b, inv_s, s255);
        v4f rc = qdq_vec4(c, inv_s, s255);
        v4f rd = qdq_vec4(d, inv_s, s255);
        __builtin_nontemporal_store(ra, &o4[i]);
        __builtin_nontemporal_store(rb, &o4[i + stride]);
        __builtin_nontemporal_store(rc, &o4[i + 2u * stride]);
        __builtin_nontemporal_store(rd, &o4[i + 3u * stride]);
    }
    // remainder float4s
    for (; i < n4; i += stride) {
        v4f v = __builtin_nontemporal_load(&x4[i]);
        __builtin_nontemporal_store(qdq_vec4(v, inv_s, s255), &o4[i]);
    }

    if (blockIdx.x == 0 && threadIdx.x == 0) {
        // scalar tail (n % 4 != 0) — N is a multiple of 4 here, but stay generic
        for (unsigned t = (n4 << 2); t < n; ++t)
            out[t] = qdq_elem(x[t], inv_s, s255);
        out[n] = s;   // second tuple output: s_x passthrough
    }
}

extern "C" void kernel_launch(void* const* d_in, const int* in_sizes, int n_in,
                              void* d_out, int out_size, void* d_ws, size_t ws_size,
                              hipStream_t stream) {
    const float* x   = (const float*)d_in[0];   // x_hat, fp32
    const float* s_x = (const float*)d_in[1];   // scale, fp32[1]
    float*       out = (float*)d_out;           // N floats + 1 float (s_x)

    const unsigned n  = (unsigned)in_sizes[0];  // 134,217,728 << 2^31
    const unsigned n4 = n >> 2;

    const int threads = 256;                    // 8 wave32s per block
    long long need = ((long long)n4 + threads - 1) / threads;
    if (need < 1) need = 1;
    const long long cap = 32768;                // at full N: exactly one 4x-unrolled iter/thread
    const int blocks = (int)(need < cap ? need : cap);

    logsqrt2_qdq_kernel<<<blocks, threads, 0, stream>>>(x, s_x, out, n);
}